// Conv1dDifferentSamplingFreq_73461120631079
// MI455X (gfx1250) — compile-verified
//
#include <hip/hip_runtime.h>

typedef __attribute__((ext_vector_type(2))) float v2f;
typedef __attribute__((ext_vector_type(8))) float v8f;

#define BATCH   64
#define NBANK   3
#define IN_CH   2
#define OUT_CH  64
#define MAX_LEN 38400
#define MAX_OUT 19200

// One wave32 computes a 16(oc) x 16(pos) output tile of
//   D[oc, p] = sum_{r < CK} W[oc, r] * X[r/K, p*2 + r%K - PAD] + bias[oc]
// via chained V_WMMA_F32_16X16X4_F32 (D = A(16x4) * B(4x16) + C).
//
// VGPR layouts per CDNA5 ISA 7.12.2 (32-bit, wave32):
//   A 16x4 : lanes 0-15 -> M=lane, VGPR{0,1}=K{0,1}; lanes 16-31 -> M=lane-16, K{2,3}
//   B 4x16 : VGPR v, lanes 0-15 -> row K=v, N=lane; lanes 16-31 -> row K=v+2, N=lane-16
//   C/D    : VGPR j, lanes 0-15 -> M=j, N=lane;     lanes 16-31 -> M=j+8,  N=lane-16
template <int K, int PAD, int LEN, int OUT_LEN>
__global__ void __launch_bounds__(256)
conv1d_wmma_kernel(const float* __restrict__ x, const float* __restrict__ W,
                   const float* __restrict__ bias, float* __restrict__ out,
                   int bank) {
  constexpr int CK = IN_CH * K;          // true reduction length
  constexpr int R  = (CK + 3) & ~3;      // padded to multiple of 4
  constexpr int PT = (OUT_LEN + 15) / 16;

  const int wave = threadIdx.x >> 5;     // wave-uniform
  const int lane = threadIdx.x & 31;

  const int wave_tile = blockIdx.x * 8 + wave;
  const int total     = BATCH * 4 * PT;
  if (wave_tile >= total) return;        // wave-uniform exit

  const int octile = wave_tile & 3;          // 4 oc-tiles of 16 channels
  const int pt     = (wave_tile >> 2) % PT;  // position tile
  const int b      = (wave_tile >> 2) / PT;  // batch

  const int  n   = lane & 15;
  const bool hiL = lane >= 16;
  const int  p   = pt * 16 + n;          // output position (B column / store col)
  const int  oc  = octile * 16 + n;      // A row for this lane

  const float* xb = x + (size_t)(b * NBANK + bank) * IN_CH * MAX_LEN;

  // Seed accumulator with bias (broadcast over positions).
  const int mrow0 = octile * 16 + (hiL ? 8 : 0);
  v8f acc;
#pragma unroll
  for (int j = 0; j < 8; ++j) acc[j] = bias[mrow0 + j];

#pragma unroll
  for (int kb = 0; kb < R / 4; ++kb) {
    const int r0 = kb * 4 + (hiL ? 2 : 0);
    v2f amat, bmat;
#pragma unroll
    for (int u = 0; u < 2; ++u) {
      const int r = r0 + u;
      float av = 0.0f, bv = 0.0f;
      if (r < CK) {                       // zero-pad reduction dim to R
        av = W[oc * CK + r];
        const int cch = r / K;            // constant div (templated K)
        const int kk  = r % K;
        const int t   = p * 2 + kk - PAD; // stride-2 conv with zero padding
        int tc = t < 0 ? 0 : (t >= LEN ? LEN - 1 : t);  // clamp: load stays unconditional
        const float xv = xb[(size_t)cch * MAX_LEN + tc];
        bv = (t == tc) ? xv : 0.0f;       // select instead of divergent branch
      }
      amat[u] = av;
      bmat[u] = bv;
    }
    acc = __builtin_amdgcn_wmma_f32_16x16x4_f32(
        /*neg_a=*/false, amat, /*neg_b=*/false, bmat,
        /*c_mod=*/(short)0, acc, /*reuse_a=*/false, /*reuse_b=*/false);
  }

  if (p < OUT_LEN) {
    float* o = out + ((size_t)(b * NBANK + bank) * OUT_CH + mrow0) * MAX_OUT + p;
#pragma unroll
    for (int j = 0; j < 8; ++j) o[(size_t)j * MAX_OUT] = acc[j];  // 64B-coalesced rows
  }
}

// Fill out[b, bank, oc, start..MAX_OUT) with NaN. Grid: (ceil(tail/256), BATCH*OUT_CH).
__global__ void __launch_bounds__(256)
nan_fill_kernel(float* __restrict__ out, int bank, int start) {
  const int row = blockIdx.y;                               // b*OUT_CH + oc
  const int p   = start + blockIdx.x * blockDim.x + threadIdx.x;
  if (p < MAX_OUT) {
    const int bb  = row >> 6;
    const int ocl = row & 63;
    out[((size_t)(bb * NBANK + bank) * OUT_CH + ocl) * MAX_OUT + p] =
        __builtin_nanf("");
  }
}

extern "C" void kernel_launch(void* const* d_in, const int* in_sizes, int n_in,
                              void* d_out, int out_size, void* d_ws, size_t ws_size,
                              hipStream_t stream) {
  const float* x  = (const float*)d_in[0];
  const float* W0 = (const float*)d_in[1];
  const float* b0 = (const float*)d_in[2];
  const float* W1 = (const float*)d_in[3];
  const float* b1 = (const float*)d_in[4];
  const float* W2 = (const float*)d_in[5];
  const float* b2 = (const float*)d_in[6];
  float* out = (float*)d_out;

  // Bank 0: K=5, pad=2, L=38400, out=19200 (covers full MAX_OUT, no NaN tail)
  {
    constexpr int PT = (19200 + 15) / 16;
    const int blocks = (BATCH * 4 * PT + 7) / 8;
    conv1d_wmma_kernel<5, 2, 38400, 19200>
        <<<blocks, 256, 0, stream>>>(x, W0, b0, out, 0);
  }
  // Bank 1: K=10, pad=5, L=19200, out=9601
  {
    constexpr int PT = (9601 + 15) / 16;
    const int blocks = (BATCH * 4 * PT + 7) / 8;
    conv1d_wmma_kernel<10, 5, 19200, 9601>
        <<<blocks, 256, 0, stream>>>(x, W1, b1, out, 1);
  }
  // Bank 2: K=20, pad=10, L=9600, out=4801
  {
    constexpr int PT = (4801 + 15) / 16;
    const int blocks = (BATCH * 4 * PT + 7) / 8;
    conv1d_wmma_kernel<20, 10, 9600, 4801>
        <<<blocks, 256, 0, stream>>>(x, W2, b2, out, 2);
  }
  // NaN tails for banks 1 and 2.
  {
    dim3 g1((MAX_OUT - 9601 + 255) / 256, BATCH * OUT_CH);
    nan_fill_kernel<<<g1, 256, 0, stream>>>(out, 1, 9601);
    dim3 g2((MAX_OUT - 4801 + 255) / 256, BATCH * OUT_CH);
    nan_fill_kernel<<<g2, 256, 0, stream>>>(out, 2, 4801);
  }
}